// MotifEnergy_32538672234671
// MI455X (gfx1250) — compile-verified
//
#include <hip/hip_runtime.h>
#include <math.h>

// ---------------------------------------------------------------------------
// MotifEnergy on gfx1250 (MI455X, wave32).
//
// Memory-bound problem (~1.07 GB reads -> ~46us @ 23.3 TB/s). Q3+K3 (51MB)
// are L2-resident (192MB L2); T (256MB) is streamed non-temporally.
// Per-motif length-64 dots are mapped onto V_WMMA_F32_16X16X4_F32:
// one wave = 16 motifs, diag of the 16x16 accumulator = the 16 dot products.
// ---------------------------------------------------------------------------

typedef __attribute__((ext_vector_type(2))) float v2f;
typedef __attribute__((ext_vector_type(8))) float v8f;

// Monotonic order-preserving float <-> uint32 key (for atomicMax on floats).
__device__ __forceinline__ unsigned fkey(float f) {
  unsigned u = __float_as_uint(f);
  return (u & 0x80000000u) ? ~u : (u | 0x80000000u);
}
__device__ __forceinline__ float fdecode(unsigned k) {
  unsigned u = (k & 0x80000000u) ? (k & 0x7fffffffu) : ~k;
  return __uint_as_float(u);
}
__device__ __forceinline__ bool finite_f(float f) {
  return (__float_as_uint(f) & 0x7f800000u) != 0x7f800000u;
}

// ---- Pass 0: zero/neutral-init workspace accumulators (ws is poisoned) ----
__global__ void init_ws_kernel(unsigned* __restrict__ nodeMaxKey,
                               float* __restrict__ nodeSum,
                               float* __restrict__ graphAcc, int N, int G) {
  int i = blockIdx.x * blockDim.x + threadIdx.x;
  if (i < N) {
    nodeMaxKey[i] = 0u;  // decodes to NaN -> "empty node" (matches reference fill)
    nodeSum[i] = 0.0f;
  }
  if (i < G) graphAcc[i] = 0.0f;
}

// ---- Pass 1: bell[m] = beta * dot/8 via WMMA; atomicMax of per-node max ----
__global__ void __launch_bounds__(256) motif_bell_kernel(
    const float* __restrict__ Q, const float* __restrict__ K,
    const float* __restrict__ T, const int* __restrict__ c3,
    const int* __restrict__ u3, const int* __restrict__ v3,
    const float* __restrict__ beta, float* __restrict__ bell,
    unsigned* __restrict__ nodeMaxKey, int M) {
  const int lane = threadIdx.x & 31;
  const int wave = (int)((blockIdx.x * blockDim.x + threadIdx.x) >> 5);
  const int m0 = wave << 4;          // 16 motifs per wave
  if (m0 >= M) return;               // wave-uniform exit: EXEC stays all-ones
  const int mLocal = lane & 15;      // A-matrix row / B-matrix column = motif
  const int m = m0 + mLocal;
  const int mc = (m < M) ? m : (M - 1);  // clamp tail loads; stores masked below
  const int ci = c3[mc];
  const float* __restrict__ qrow  = Q + (size_t)ci * 64;
  const float* __restrict__ kurow = K + (size_t)u3[mc] * 64;
  const float* __restrict__ kvrow = K + (size_t)v3[mc] * 64;
  const float* __restrict__ trow  = T + (size_t)mc * 64;

  // ISA layouts for f32 16x16x4:
  //   A (16x4): lanes 0-15 hold K=k0+{0,1}; lanes 16-31 hold K=k0+{2,3}
  //   B (4x16): VGPR0 = row K=k0 (lanes 0-15) / K=k0+2 (lanes 16-31), VGPR1 = +1
  // => identical per-lane K offsets for A and B.
  const int koff = (lane >> 4) << 1;  // 0 or 2

  v8f acc = {};
#pragma unroll
  for (int k0 = 0; k0 < 64; k0 += 4) {
    const int kb = k0 + koff;
    v2f a  = *(const v2f*)(qrow + kb);
    v2f ku = *(const v2f*)(kurow + kb);
    v2f kv = *(const v2f*)(kvrow + kb);
    v2f t  = __builtin_nontemporal_load((const v2f*)(trow + kb));  // streamed
    v2f b  = ku * kv + t;
    // D = A x B + C ; diag(D) accumulates the 16 motif dot products.
    acc = __builtin_amdgcn_wmma_f32_16x16x4_f32(
        /*neg_a=*/false, a, /*neg_b=*/false, b,
        /*c_mod=*/(short)0, acc, /*reuse_a=*/false, /*reuse_b=*/false);
  }

  // Diagonal extraction (C/D layout: lane<16 -> M=vgpr,N=lane;
  // lane>=16 -> M=vgpr+8,N=lane-16). (m,m) lives at:
  //   lanes 0-7  : acc[lane]      (m = lane)
  //   lanes 24-31: acc[lane-24]   (m = lane-16)
  const int rr = (lane < 16) ? lane : (lane - 24);
  const int r = rr & 7;
  float dv = acc[0];
  if (r == 1) dv = acc[1];
  if (r == 2) dv = acc[2];
  if (r == 3) dv = acc[3];
  if (r == 4) dv = acc[4];
  if (r == 5) dv = acc[5];
  if (r == 6) dv = acc[6];
  if (r == 7) dv = acc[7];

  if (rr >= 0 && rr < 8 && m < M) {
    const float bv = beta[0] * dv * 0.125f;  // scale = sqrt(1*64) = 8
    __builtin_nontemporal_store(bv, bell + m);
    atomicMax(nodeMaxKey + ci, fkey(bv));
  }
}

// ---- Pass 2: nodeSum[c] += exp(bell - max_safe[c]) ----
__global__ void expsum_kernel(const float* __restrict__ bell,
                              const int* __restrict__ c3,
                              const unsigned* __restrict__ nodeMaxKey,
                              float* __restrict__ nodeSum, int M) {
  int m = blockIdx.x * blockDim.x + threadIdx.x;
  if (m >= M) return;
  const int ci = c3[m];
  const float mx = fdecode(nodeMaxKey[ci]);
  const float msafe = finite_f(mx) ? mx : 0.0f;
  atomicAdd(nodeSum + ci, expf(bell[m] - msafe));
}

// ---- Pass 3: per-node lse -> per-graph accumulation ----
__global__ void node_to_graph_kernel(const float* __restrict__ nodeSum,
                                     const unsigned* __restrict__ nodeMaxKey,
                                     const int* __restrict__ batch,
                                     float* __restrict__ graphAcc, int N) {
  int n = blockIdx.x * blockDim.x + threadIdx.x;
  if (n >= N) return;
  const float s = nodeSum[n];
  const float mx = fdecode(nodeMaxKey[n]);
  const float msafe = finite_f(mx) ? mx : 0.0f;
  const float lse = (s > 0.0f) ? (logf(s) + msafe) : 0.0f;
  atomicAdd(graphAcc + batch[n], lse);
}

// ---- Pass 4: out = (lambda/beta) * graphAcc ----
__global__ void finalize_kernel(const float* __restrict__ graphAcc,
                                const float* __restrict__ lam,
                                const float* __restrict__ beta,
                                float* __restrict__ out, int G) {
  int g = blockIdx.x * blockDim.x + threadIdx.x;
  if (g < G) out[g] = (lam[0] / beta[0]) * graphAcc[g];
}

extern "C" void kernel_launch(void* const* d_in, const int* in_sizes, int n_in,
                              void* d_out, int out_size, void* d_ws, size_t ws_size,
                              hipStream_t stream) {
  const float* Q    = (const float*)d_in[0];  // [N,1,64]
  const float* K    = (const float*)d_in[1];  // [N,1,64]
  const float* T    = (const float*)d_in[2];  // [M,1,64]
  const float* lam  = (const float*)d_in[3];  // [1]
  const float* beta = (const float*)d_in[4];  // [1]
  const int* c3     = (const int*)d_in[5];    // [M]
  const int* u3     = (const int*)d_in[6];    // [M]
  const int* v3     = (const int*)d_in[7];    // [M]
  const int* batch  = (const int*)d_in[8];    // [N]

  const int M = in_sizes[5];
  const int N = in_sizes[8];
  const int G = out_size;

  // Workspace layout (all 4-byte aligned): bell[M] | maxKey[N] | sum[N] | graph[G]
  float* bell = (float*)d_ws;
  unsigned* nodeMaxKey = (unsigned*)(bell + M);
  float* nodeSum = (float*)(nodeMaxKey + N);
  float* graphAcc = nodeSum + N;

  {
    int n = (N > G) ? N : G;
    init_ws_kernel<<<(n + 255) / 256, 256, 0, stream>>>(nodeMaxKey, nodeSum,
                                                        graphAcc, N, G);
  }
  {
    long long waves = ((long long)M + 15) / 16;   // 16 motifs per wave32
    long long threads = waves * 32;
    int blocks = (int)((threads + 255) / 256);
    motif_bell_kernel<<<blocks, 256, 0, stream>>>(Q, K, T, c3, u3, v3, beta,
                                                  bell, nodeMaxKey, M);
  }
  expsum_kernel<<<(M + 255) / 256, 256, 0, stream>>>(bell, c3, nodeMaxKey,
                                                     nodeSum, M);
  node_to_graph_kernel<<<(N + 255) / 256, 256, 0, stream>>>(nodeSum, nodeMaxKey,
                                                            batch, graphAcc, N);
  finalize_kernel<<<(G + 255) / 256, 256, 0, stream>>>(graphAcc, lam, beta,
                                                       (float*)d_out, G);
}